// SSKernel_2619930050872
// MI455X (gfx1250) — compile-verified
//
#include <hip/hip_runtime.h>
#include <hip/hip_bf16.h>
#include <math.h>

#define HH   512
#define NCC  32
#define LL   8192

typedef __attribute__((ext_vector_type(2))) float v2f;
typedef __attribute__((ext_vector_type(8))) float v8f;

#define LOG2E_F   1.4426950408889634f
#define INV2PI_D  0.15915494309189535  /* 1/(2*pi), double */
#define INV2PI_F  0.15915494f

#define NA 2    /* A fragments (super-tiles) per wave */
#define NB 4    /* B fragments (column groups) per wave */

// Factorization: l = lA + lB with
//   lB = j + 16*c          (c = 0..3: four B fragments P_c[n,j] = exp(dtA*lB))
//   lA = 64*(16*ta + ii)   (ta: super-tile; A fragment D[n,ii] = C2*exp(dtA*lA))
// Tile (ta,c) is a 16x16 GEMM over r = 2n+part (64 reals, 4 per
// V_WMMA_F32_16X16X4_F32). Each wave owns ta in {2wh, 2wh+1} x c in 0..3:
// per chunk it builds 2 A + 4 B fragments and issues 8 WMMAs (2048 outputs),
// amortizing the transcendentals on BOTH operands.
//
// Block (256 thr = 8 waves) covers 2 heads (4 waves = 8192 outputs per head).
// Phases are in *revolutions* (v_sin/v_cos native domain). B phases reach
// ~97 rev (f32-exact feed, inside v_sin's valid range); A revolution counts
// (up to ~1.3e4) are reduced exactly with one f64 mul+floor — no libm
// Payne-Hanek slow path anywhere.
__global__ __launch_bounds__(256)
void s4d_wmma_kernel(const float* __restrict__ log_dt,
                     const float* __restrict__ log_A_real,
                     const float* __restrict__ A_imag,
                     const float* __restrict__ B_ri,
                     const float* __restrict__ C_ri,
                     float* __restrict__ K)
{
    __shared__ float  s_ex [2][NCC];   // dtA.re * log2(e)      (for v_exp_f32)
    __shared__ float  s_ph [2][NCC];   // dtA.im / (2*pi), f32  (B phases)
    __shared__ double s_phd[2][NCC];   // dtA.im / (2*pi), f64  (A phases)
    __shared__ float2 s_C2 [2][NCC];   // 2 * C * B * (exp(dtA)-1)/A

    const int tid = threadIdx.x;

    // ---- per-head mode precompute: 64 lanes cover 2 heads x 32 modes ----
    if (tid < 2 * NCC) {
        const int hh = tid >> 5;                   // head slot in block
        const int n  = tid & 31;
        const int h  = 2 * blockIdx.x + hh;
        const int hn = h * NCC + n;
        const float dt = __expf(log_dt[h]);
        const float Ar = -__expf(log_A_real[hn]);
        const float Ai = A_imag[hn];
        const float dr = Ar * dt;
        const float di = Ai * dt;                  // >= 0
        const float Br = B_ri[2 * hn + 0], Bi = B_ri[2 * hn + 1];
        const float Cr = C_ri[2 * hn + 0], Ci = C_ri[2 * hn + 1];   // CH == 1
        // exp(dtA) - 1  (di < ~1 rad: fast path fine)
        float sw, cw;
        __sincosf(di, &sw, &cw);
        const float er = __expf(dr);
        const float nr = er * cw - 1.0f;
        const float ni = er * sw;
        // (exp(dtA)-1)/A
        const float inv = 1.0f / (Ar * Ar + Ai * Ai);
        const float fr  = (nr * Ar + ni * Ai) * inv;
        const float fi  = (ni * Ar - nr * Ai) * inv;
        // C*B
        const float cbr = Cr * Br - Ci * Bi;
        const float cbi = Cr * Bi + Ci * Br;

        s_ex [hh][n] = dr * LOG2E_F;
        s_ph [hh][n] = di * INV2PI_F;
        s_phd[hh][n] = (double)di * INV2PI_D;
        s_C2 [hh][n] = make_float2(2.0f * (cbr * fr - cbi * fi),
                                   2.0f * (cbr * fi + cbi * fr));
    }
    __syncthreads();

    // ---- per-wave subtile grid ----
    const int wave = tid >> 5;
    const int lane = tid & 31;
    const int hi   = lane >> 4;        // lane-half: mode parity / D rows M+8
    const int low  = lane & 15;        // A row ii / B col j
    const int hh   = wave >> 2;        // head slot (0/1) in block
    const int wh   = wave & 3;         // wave within head
    const int h    = 2 * blockIdx.x + hh;

    // B exponents lB = j + 16c
    float fB[NB];
#pragma unroll
    for (int c = 0; c < NB; ++c)
        fB[c] = (float)(low + 16 * c);

    // A exponents lA = 1024*ta + 64*ii, ta = 2*wh + aa
    float  fA [NA];
    double fAd[NA];
#pragma unroll
    for (int aa = 0; aa < NA; ++aa) {
        const int lA = 1024 * (2 * wh + aa) + 64 * low;
        fA [aa] = (float)lA;
        fAd[aa] = (double)lA;
    }

    v8f acc[NA][NB];
#pragma unroll
    for (int aa = 0; aa < NA; ++aa)
#pragma unroll
        for (int c = 0; c < NB; ++c)
            acc[aa][c] = (v8f){0.f, 0.f, 0.f, 0.f, 0.f, 0.f, 0.f, 0.f};

#pragma unroll
    for (int kk = 0; kk < 16; ++kk) {
        const int n = 2 * kk + hi;                 // this lane's mode this chunk
        const float  k2 = s_ex [hh][n];
        const float  pf = s_ph [hh][n];
        const double pd = s_phd[hh][n];
        const float2 c2 = s_C2 [hh][n];

        // 4 B fragments: P_c[n,j] = exp(dtA*(j+16c)), phases <= ~97 rev.
        v2f Bf[NB];
#pragma unroll
        for (int c = 0; c < NB; ++c) {
            const float eb = __builtin_amdgcn_exp2f(k2 * fB[c]);
            const float ub = pf * fB[c];
            Bf[c].x = eb * __builtin_amdgcn_cosf(ub);
            Bf[c].y = eb * __builtin_amdgcn_sinf(ub);
        }

        // 2 A fragments: D[n,ii] = C2[n]*exp(dtA*lA), exact f64 reduction.
        v2f Af[NA];
#pragma unroll
        for (int aa = 0; aa < NA; ++aa) {
            const double ta = pd * fAd[aa];             // revolutions (>= 0)
            const float  ua = (float)(ta - floor(ta));  // frac -> [0,1)
            const float  ea = __builtin_amdgcn_exp2f(k2 * fA[aa]);
            const float  sa = __builtin_amdgcn_sinf(ua);
            const float  ca = __builtin_amdgcn_cosf(ua);
            const float  qr = ea * ca, qi = ea * sa;
            Af[aa].x =  c2.x * qr - c2.y * qi;          //  Re(D)
            Af[aa].y = -(c2.x * qi + c2.y * qr);        // -Im(D)
        }

        // 8 WMMAs, 8 independent accumulator chains
#pragma unroll
        for (int aa = 0; aa < NA; ++aa)
#pragma unroll
            for (int c = 0; c < NB; ++c)
                acc[aa][c] = __builtin_amdgcn_wmma_f32_16x16x4_f32(
                    false, Af[aa], false, Bf[c], (short)0, acc[aa][c],
                    false, false);
    }

    // ---- store: (ta, c) tile, element (M = r + 8*hi, N = low)
    //      -> l = 1024*ta + 64*M + 16*c + low ----
    float* outh = K + (size_t)h * LL;
#pragma unroll
    for (int aa = 0; aa < NA; ++aa) {
        float* out = outh + 1024 * (2 * wh + aa);
#pragma unroll
        for (int c = 0; c < NB; ++c) {
#pragma unroll
            for (int r = 0; r < 8; ++r) {
                out[64 * (r + 8 * hi) + 16 * c + low] = acc[aa][c][r];
            }
        }
    }
}

extern "C" void kernel_launch(void* const* d_in, const int* in_sizes, int n_in,
                              void* d_out, int out_size, void* d_ws, size_t ws_size,
                              hipStream_t stream) {
    const float* log_dt     = (const float*)d_in[0];
    const float* log_A_real = (const float*)d_in[1];
    const float* A_imag     = (const float*)d_in[2];
    const float* B_ri       = (const float*)d_in[3];
    const float* C_ri       = (const float*)d_in[4];
    // d_in[5] is L (scalar) — fixed at 8192 here.
    float* K = (float*)d_out;

    dim3 grid(HH / 2);   // 2 heads per block; 4 waves/head; 8 subtiles/wave
    dim3 block(256);
    s4d_wmma_kernel<<<grid, block, 0, stream>>>(log_dt, log_A_real, A_imag,
                                                B_ri, C_ri, K);
}